// CenterLossLayer_44195213476627
// MI455X (gfx1250) — compile-verified
//
#include <hip/hip_runtime.h>

// CenterLoss for MI455X (gfx1250).
// B=256 samples, C=50000 classes, D=512 dims. ALPHA=0.5.
// Memory-bound problem: ~256 MB of unavoidable traffic -> ~11us at 23.3 TB/s.
// The matmuls in the reference are a disguised gather/scatter over a one-hot.

#define B_N 256
#define C_N 50000
#define D_N 512
#define ALPHA_F 0.5f

typedef __attribute__((ext_vector_type(2))) float v2f;
typedef __attribute__((ext_vector_type(4))) float v4f;
typedef __attribute__((ext_vector_type(8))) float v8f;

// ---------------------------------------------------------------------------
// Kernel 1: recover labels from the one-hot matrix.
// 256 blocks (one per sample), 256 threads scan 50000 floats (12500 float4).
// 51.2 MB streamed once -> nontemporal loads (TH=NT), never re-read.
// ---------------------------------------------------------------------------
__global__ void __launch_bounds__(256)
find_labels_kernel(const float* __restrict__ onehot, int* __restrict__ labels) {
    const int b = blockIdx.x;
    const v4f* row = (const v4f*)(onehot + (size_t)b * C_N);
    for (int c4 = threadIdx.x; c4 < C_N / 4; c4 += 256) {
        v4f v = __builtin_nontemporal_load(row + c4);
        // exactly one nonzero per row -> at most one thread writes
        if (v.x != 0.0f) labels[b] = 4 * c4 + 0;
        if (v.y != 0.0f) labels[b] = 4 * c4 + 1;
        if (v.z != 0.0f) labels[b] = 4 * c4 + 2;
        if (v.w != 0.0f) labels[b] = 4 * c4 + 3;
    }
}

// ---------------------------------------------------------------------------
// Kernel 2: per-sample class multiplicity (== onehot.T row-sum for its class).
// One block, labels staged in LDS; no global atomics, no 50000-entry zeroing.
// ---------------------------------------------------------------------------
__global__ void __launch_bounds__(B_N)
count_labels_kernel(const int* __restrict__ labels, int* __restrict__ cnt) {
    __shared__ int sl[B_N];
    const int b = threadIdx.x;
    sl[b] = labels[b];
    __syncthreads();
    const int lb = sl[b];
    int c = 0;
#pragma unroll 8
    for (int j = 0; j < B_N; ++j) c += (sl[j] == lb) ? 1 : 0;
    cnt[b] = c;
}

// ---------------------------------------------------------------------------
// Kernel 3: the bandwidth kernel. new_centers = centers for all 50000 rows
// (untouched classes have delta == 0/(0+1) == 0). 102.4 MB -> 102.4 MB as
// b128 nontemporal load/store pairs. 6.4M float4 = 25000 blocks x 256 lanes.
// ---------------------------------------------------------------------------
__global__ void __launch_bounds__(256)
copy_centers_kernel(const float* __restrict__ src, float* __restrict__ dst) {
    const size_t i = (size_t)blockIdx.x * 256 + threadIdx.x; // float4 units
    if (i < (size_t)C_N * D_N / 4) {
        const v4f* s = (const v4f*)src;
        v4f* d = (v4f*)dst;
        v4f v = __builtin_nontemporal_load(s + i);
        __builtin_nontemporal_store(v, d + i);
    }
}

// ---------------------------------------------------------------------------
// Kernel 4: fix up the <=256 touched rows of new_centers.
// Each sample b contributes -ALPHA/(cnt+1) * (centers[lb] - x[b]) to row lb;
// the coefficient depends only on the class, so per-sample global f32 atomics
// sum to exactly the reference's delta. 131072 atomics total (tiny).
// ---------------------------------------------------------------------------
__global__ void __launch_bounds__(256)
scatter_adjust_kernel(const float* __restrict__ x, const float* __restrict__ centers,
                      const int* __restrict__ labels, const int* __restrict__ cnt,
                      float* __restrict__ new_centers) {
    const int b = blockIdx.x;
    const int lb = labels[b];
    const float coef = -ALPHA_F / (float)(cnt[b] + 1);
    const float* cr = centers + (size_t)lb * D_N;
    const float* xr = x + (size_t)b * D_N;
    float* nr = new_centers + (size_t)lb * D_N;
    for (int d = threadIdx.x; d < D_N; d += 256) {
        atomicAdd(&nr[d], coef * (cr[d] - xr[d]));
    }
}

// ---------------------------------------------------------------------------
// Kernel 5: result[b] = ||x[b] - centers[label[b]]||^2 via V_WMMA_F32_16X16X4_F32.
// One wave32 per block handles 16 samples.
//
// A-matrix (16x4 f32) layout: lane L holds row M=L&15; VGPR0/1 hold
// K={0,1} for lanes 0-15 and K={2,3} for lanes 16-31. The B-matrix (4x16)
// layout B[k][n] with rows striped across lanes puts B = A^T in the *same*
// register image when B's data is diff^T, so wmma(a, a, acc) accumulates the
// 16x16 Gram matrix; its diagonal after 128 K-steps (K=512) is the answer.
// EXEC is all-ones throughout the WMMA loop (divergence only at final store).
// ---------------------------------------------------------------------------
__global__ void __launch_bounds__(32)
result_wmma_kernel(const float* __restrict__ x, const float* __restrict__ centers,
                   const int* __restrict__ labels, float* __restrict__ result) {
    const int lane = threadIdx.x;        // 0..31
    const int m = lane & 15;             // matrix row this lane feeds
    const int kb2 = lane >> 4;           // v2f group within each K=4 step
    const int row = blockIdx.x * 16 + m; // sample index
    const int lb = labels[row];

    const v2f* xr = (const v2f*)(x + (size_t)row * D_N);
    const v2f* cr = (const v2f*)(centers + (size_t)lb * D_N);

    v8f acc = {};
#pragma unroll 4
    for (int k2 = 0; k2 < D_N / 2; k2 += 2) {   // K advances by 4 per wmma
        v2f a = xr[k2 + kb2] - cr[k2 + kb2];    // diff slice: A and B=A^T
        acc = __builtin_amdgcn_wmma_f32_16x16x4_f32(
            /*neg_a=*/false, a, /*neg_b=*/false, a,
            /*c_mod=*/(short)0, acc, /*reuse_a=*/false, /*reuse_b=*/false);
    }

    // Diagonal extraction per 16x16 f32 C/D layout:
    //   lanes 0-15 hold (M=v, N=lane), lanes 16-31 hold (M=v+8, N=lane-16).
    //   (m,m) for m<8  -> lane m,     vgpr m
    //   (m,m) for m>=8 -> lane m+16,  vgpr m-8
    const bool active = (lane < 8) || (lane >= 24);
    const int vi = (lane < 8) ? lane : (lane >= 24 ? lane - 24 : 0);
    float r = 0.0f;
#pragma unroll
    for (int i = 0; i < 8; ++i)
        if (vi == i) r = acc[i];
    if (active) {
        const int out_m = (lane < 8) ? lane : (lane - 16);
        result[blockIdx.x * 16 + out_m] = r;
    }
}

// ---------------------------------------------------------------------------
extern "C" void kernel_launch(void* const* d_in, const int* in_sizes, int n_in,
                              void* d_out, int out_size, void* d_ws, size_t ws_size,
                              hipStream_t stream) {
    const float* x       = (const float*)d_in[0]; // [256, 512]
    const float* onehot  = (const float*)d_in[1]; // [256, 50000]
    const float* centers = (const float*)d_in[2]; // [50000, 512]

    float* result      = (float*)d_out;           // [256] (output 0)
    float* new_centers = (float*)d_out + B_N;     // [50000, 512] (output 1)

    int* labels = (int*)d_ws;        // 256 ints, rewritten every call
    int* cnt    = labels + B_N;      // 256 ints, rewritten every call

    find_labels_kernel<<<B_N, 256, 0, stream>>>(onehot, labels);
    count_labels_kernel<<<1, B_N, 0, stream>>>(labels, cnt);

    const int copy_blocks = (int)(((size_t)C_N * D_N / 4 + 255) / 256); // 25000
    copy_centers_kernel<<<copy_blocks, 256, 0, stream>>>(centers, new_centers);

    scatter_adjust_kernel<<<B_N, 256, 0, stream>>>(x, centers, labels, cnt, new_centers);
    result_wmma_kernel<<<B_N / 16, 32, 0, stream>>>(x, centers, labels, result);
}